// Invert2_10_20684562497569
// MI455X (gfx1250) — compile-verified
//
#include <hip/hip_runtime.h>

// Invert2_10: 16-step spiking-threshold recurrence, elementwise over fp32.
// Memory-roofline streaming kernel for MI455X (gfx1250, wave32).
//
// Algebraic simplification: (v - T)/(|v| + 1) > 0  <=>  v > T, since the
// denominator is always >= 1 (positive). This removes all 16 divides per
// element, leaving ~4 VALU ops per step -> kernel is HBM-bandwidth-bound
// (536 MB / 23.3 TB/s ~= 23 us floor).

typedef __attribute__((ext_vector_type(4))) float v4f;

__device__ __forceinline__ float invert_step16(float x) {
    constexpr float H[16] = {
        -0.89521986f, 1.8242345f, 1.0888329f, 1.2783841f,
         1.1793505f,  5.402602f,  3.556659f, -0.45232677f,
         1.8296803f,  2.5809617f, 4.8217273f, 2.7070105f,
        -0.76238555f, 3.3387434f, 1.7290733f, 5.140637f};
    constexpr float D[16] = {
        -0.18134391f, -0.07921064f, -0.04980344f, -0.02413579f,
         0.50927377f, -0.0713518f,  -0.64178044f, -0.02694374f,
        -0.06014722f, -0.0580623f,  -1.1322998f,  -0.46733513f,
        -0.05247239f, -0.04035916f, -0.02469978f, -0.01966147f};
    constexpr float T[16] = {
         2.821549f,   2.1506927f,  1.9763926f,  2.5215197f,
        -0.99999964f,-1.4365319f,  0.9995256f, -1.347566f,
        -2.1773431f, -2.2306669f, -1.4854108f,  0.9888728f,
        -2.9239423f, -3.8899376f, -3.105946f,  -4.132591f};

    // jnp.sign semantics: sign(0) == 0
    float s = (x > 0.0f) ? 1.0f : ((x < 0.0f) ? -1.0f : 0.0f);
    float v = __builtin_fabsf(x);
    float z = 0.0f;
    float o = 0.0f;
#pragma unroll
    for (int t = 0; t < 16; ++t) {
        v = __builtin_fmaf(-H[t], z, v);      // v -= z * H[t]
        z = (v > T[t]) ? 1.0f : 0.0f;         // sign of (v-T)/(|v|+1), den > 0
        o = __builtin_fmaf(D[t], z, o);       // out += z * D[t]
    }
    return o * s;
}

__global__ __launch_bounds__(256) void invert2_10_kernel(
    const float* __restrict__ x, float* __restrict__ out, int n4, int n) {
    int i = (int)blockIdx.x * 256 + (int)threadIdx.x;
    if (i < n4) {
        // b128 nontemporal load: streaming set (512 MB) has zero reuse and
        // exceeds the 192 MB L2 -> NT temporal hint avoids cache pollution.
        v4f xv = __builtin_nontemporal_load((const v4f*)x + i);
        v4f r;
#pragma unroll
        for (int c = 0; c < 4; ++c) r[c] = invert_step16(xv[c]);
        __builtin_nontemporal_store(r, (v4f*)out + i);
    }
    // Scalar tail for n not divisible by 4 (defensive; n = 4*4096*4096 here).
    int tail = n - n4 * 4;
    if (blockIdx.x == 0 && (int)threadIdx.x < tail) {
        int j = n4 * 4 + (int)threadIdx.x;
        out[j] = invert_step16(x[j]);
    }
}

extern "C" void kernel_launch(void* const* d_in, const int* in_sizes, int n_in,
                              void* d_out, int out_size, void* d_ws, size_t ws_size,
                              hipStream_t stream) {
    (void)n_in; (void)out_size; (void)d_ws; (void)ws_size;
    const float* x = (const float*)d_in[0];
    float* out = (float*)d_out;
    int n  = in_sizes[0];       // 4*4096*4096 = 67,108,864
    int n4 = n / 4;             // 16,777,216 float4 elements
    int blocks = (n4 + 255) / 256;
    if (blocks < 1) blocks = 1;
    invert2_10_kernel<<<blocks, 256, 0, stream>>>(x, out, n4, n);
}